// PINNFlow_27273042329910
// MI455X (gfx1250) — compile-verified
//
#include <hip/hip_runtime.h>
#include <cmath>

typedef float v8f __attribute__((ext_vector_type(8)));
typedef float v2f __attribute__((ext_vector_type(2)));

#define HID 64
#define W2PAD 72   // row stride for W2 in LDS: 2*72 mod 64 = 16 -> half-waves hit disjoint banks
#define H1PAD 17   // stride for h1 cache: conflict-free for both half-waves
#define WAVES 4
#define PTSW 16    // points per wave
#define PTSB (WAVES*PTSW)

// Branch-free tanh: hardware V_TANH_F32 on gfx1250, else exp2/rcp fallback.
__device__ __forceinline__ float fast_tanh(float x) {
#if __has_builtin(__builtin_amdgcn_tanhf)
    return __builtin_amdgcn_tanhf(x);
#else
    const float ax = __builtin_fabsf(x);
    const float e  = __builtin_amdgcn_exp2f(-2.885390081777927f * ax); // exp(-2|x|)
    const float r  = __builtin_amdgcn_rcpf(1.f + e);
    const float t  = __builtin_fmaf(-2.f * e, r, 1.f);                 // 1 - 2e/(1+e)
    return __builtin_copysignf(t, x);
#endif
}

__global__ __launch_bounds__(128) void pinn_jet_wmma_kernel(
    const float* __restrict__ phi, const float* __restrict__ theta,
    const float* __restrict__ radius, const float* __restrict__ hth,
    const float* __restrict__ hph, const float* __restrict__ brr,
    const float* __restrict__ W1, const float* __restrict__ b1,
    const float* __restrict__ W2, const float* __restrict__ b2,
    const float* __restrict__ W3,
    float* __restrict__ out, int N)
{
    __shared__ float w2s[HID][W2PAD];          // W2[h_in][h_out]
    __shared__ float w1a[HID], w1b[HID], b1s[HID], b2s[HID], w3s[HID];
    __shared__ float h1s[WAVES][HID][H1PAD];   // h1s[w][h][pt]

    const int tid  = threadIdx.x;
    const int wave = tid >> 5;
    const int lane = tid & 31;
    const int g    = lane >> 4;   // half-wave (selects K pair / M+8)
    const int n    = lane & 15;   // point within tile (B/D column), also A row

    // ---- phase 0: stage weights into LDS ----
    for (int idx = tid; idx < HID * HID; idx += blockDim.x)
        w2s[idx >> 6][idx & 63] = W2[idx];
    if (tid < HID) {
        w1a[tid] = W1[tid];
        w1b[tid] = W1[HID + tid];
        b1s[tid] = b1[tid];
        b2s[tid] = b2[tid];
        w3s[tid] = W3[tid];
    }
    __syncthreads();

    const int base = blockIdx.x * PTSB + wave * PTSW;

    // ---- phase 1: layer 1 tanh, cached per wave ----
    {
        const float p = phi[base + n];
        const float t = theta[base + n];
        const int h0 = g * 32;
        for (int h = h0; h < h0 + 32; ++h) {
            float z = fmaf(p, w1a[h], fmaf(t, w1b[h], b1s[h]));
            h1s[wave][h][n] = fast_tanh(z);
        }
    }
    __syncthreads();

    // T-derivative accumulators: Tp, Tt, Tpp, Tpt, Ttt, Tppp, Tppt, Tptt, Tttt
    float Tacc[9];
#pragma unroll
    for (int c = 0; c < 9; ++c) Tacc[c] = 0.f;

    const int m = (lane & 15);  // A-fragment row within tile (h_out - 16*mt)

    for (int mt = 0; mt < 4; ++mt) {
        v8f acc[10];
#pragma unroll
        for (int c = 0; c < 10; ++c)
#pragma unroll
            for (int v = 0; v < 8; ++v) acc[c][v] = 0.f;
        // bias only feeds the value channel
#pragma unroll
        for (int v = 0; v < 8; ++v) acc[0][v] = b2s[mt * 16 + g * 8 + v];

#pragma unroll 2
        for (int kt = 0; kt < 16; ++kt) {
            const int k0 = kt * 4 + g * 2;      // this lane's K pair (h_in)
            // A fragment: W2^T[16*mt + m][k0, k0+1] = W2[k][h_out]
            v2f a;
            a.x = w2s[k0][mt * 16 + m];
            a.y = w2s[k0 + 1][mt * 16 + m];
            // B fragments: 10 jet channels of h1 at (point n, h_in = k0, k0+1)
            v2f bf[10];
#pragma unroll
            for (int j = 0; j < 2; ++j) {
                const int hh = k0 + j;
                const float A  = w1a[hh];
                const float B  = w1b[hh];
                const float hv = h1s[wave][hh][n];
                const float h2 = hv * hv;
                const float g1 = 1.f - h2;                     // tanh'
                const float g2 = -2.f * hv * g1;               // tanh''
                const float g3 = -2.f * g1 * (1.f - 3.f * h2); // tanh'''
                const float A2 = A * A, AB = A * B, B2 = B * B;
                bf[0][j] = hv;
                bf[1][j] = g1 * A;
                bf[2][j] = g1 * B;
                bf[3][j] = g2 * A2;
                bf[4][j] = g2 * AB;
                bf[5][j] = g2 * B2;
                bf[6][j] = g3 * A2 * A;
                bf[7][j] = g3 * A2 * B;
                bf[8][j] = g3 * A * B2;
                bf[9][j] = g3 * B2 * B;
            }
#pragma unroll
            for (int c = 0; c < 10; ++c)
                acc[c] = __builtin_amdgcn_wmma_f32_16x16x4_f32(
                    false, a, false, bf[c], (short)0, acc[c], false, false);
        }

        // epilogue for this h_out tile: Faa di Bruno through tanh + W3 partial dot
#pragma unroll
        for (int v = 0; v < 8; ++v) {
            const float z0 = acc[0][v], zp = acc[1][v], zt = acc[2][v];
            const float zpp = acc[3][v], zpt = acc[4][v], ztt = acc[5][v];
            const float zppp = acc[6][v], zppt = acc[7][v], zptt = acc[8][v], zttt = acc[9][v];
            const float u  = fast_tanh(z0);
            const float u2 = u * u;
            const float f1 = 1.f - u2;
            const float f2 = -2.f * u * f1;
            const float f3 = -2.f * f1 * (1.f - 3.f * u2);
            const float up   = f1 * zp;
            const float ut   = f1 * zt;
            const float upp  = f2 * zp * zp + f1 * zpp;
            const float upt  = f2 * zp * zt + f1 * zpt;
            const float utt  = f2 * zt * zt + f1 * ztt;
            const float uppp = f3 * zp * zp * zp + 3.f * f2 * zp * zpp + f1 * zppp;
            const float uppt = f3 * zp * zp * zt + f2 * (zpp * zt + 2.f * zp * zpt) + f1 * zppt;
            const float uptt = f3 * zp * zt * zt + f2 * (ztt * zp + 2.f * zt * zpt) + f1 * zptt;
            const float uttt = f3 * zt * zt * zt + 3.f * f2 * zt * ztt + f1 * zttt;
            const float w3h = w3s[mt * 16 + g * 8 + v];
            Tacc[0] = fmaf(w3h, up,   Tacc[0]);
            Tacc[1] = fmaf(w3h, ut,   Tacc[1]);
            Tacc[2] = fmaf(w3h, upp,  Tacc[2]);
            Tacc[3] = fmaf(w3h, upt,  Tacc[3]);
            Tacc[4] = fmaf(w3h, utt,  Tacc[4]);
            Tacc[5] = fmaf(w3h, uppp, Tacc[5]);
            Tacc[6] = fmaf(w3h, uppt, Tacc[6]);
            Tacc[7] = fmaf(w3h, uptt, Tacc[7]);
            Tacc[8] = fmaf(w3h, uttt, Tacc[8]);
        }
    }

    // combine the two half-wave partial sums (h_out v vs v+8)
#pragma unroll
    for (int c = 0; c < 9; ++c)
        Tacc[c] += __shfl_xor(Tacc[c], 16, 32);

    if (lane < 16) {
        const int gi = base + n;
        const float Tp = Tacc[0], Tt = Tacc[1], Tpp = Tacc[2], Tpt = Tacc[3], Ttt = Tacc[4];
        const float Tppp = Tacc[5], Tppt = Tacc[6], Tptt = Tacc[7], Tttt = Tacc[8];
        const float t  = theta[gi];
        const float r  = radius[gi];
        const float h_t = hth[gi];
        const float h_p = hph[gi];
        const float bv  = brr[gi];
        const float s = sinf(t), c = cosf(t);
        const float invs = 1.f / s;
        const float ut_ = Tp * invs + Tt;           // u_theta
        const float up_ = Tp * invs - Tt;           // u_phi
        const float c2ms2 = c * c - s * s;
        const float uthT  = -(Tpt + Ttt * s + Tt * c);                                  // u_theta_theta
        const float uthTT = -(Tptt * s + Tpt * c + Tttt * s * s + 3.f * Ttt * s * c + Tt * c2ms2);
        const float uthPP = -(Tppp * invs + Tppt);                                      // u_theta_phi_phi
        const float uphTT =  (Tptt * s + Tpt * c - Tttt * s * s - 3.f * Ttt * s * c - Tt * c2ms2);
        const float uphP  = Tpp * invs - Tpt;                                           // u_phi_phi
        const float uphPP = Tppp * invs - Tppt;                                         // u_phi_phi_phi
        const float invr = 1.f / r;
        const float invrs = invr * invs;
        const float div_uh = (uthT + uphP) * invrs;
        const float lapT = s * invr * invr * uthTT + uthPP * invrs * invrs;
        const float lapP = s * invr * invr * uphTT + uphPP * invrs * invrs;
        const float comp = s * (lapT * lapT + lapP * lapP);
        const float sv   = -(ut_ * h_t + up_ * h_p) - bv * div_uh;
        const float tg   = div_uh - ut_ * (s / c) * invr;
        out[gi]         = ut_;
        out[N + gi]     = up_;
        out[2 * N + gi] = div_uh;
        out[3 * N + gi] = sv;
        out[4 * N + gi] = tg;
        out[5 * N + gi] = comp;
    }
}

extern "C" void kernel_launch(void* const* d_in, const int* in_sizes, int n_in,
                              void* d_out, int out_size, void* d_ws, size_t ws_size,
                              hipStream_t stream) {
    const float* phi    = (const float*)d_in[0];
    const float* theta  = (const float*)d_in[1];
    const float* radius = (const float*)d_in[2];
    const float* hth    = (const float*)d_in[3];
    const float* hph    = (const float*)d_in[4];
    const float* brr    = (const float*)d_in[5];
    const float* W1     = (const float*)d_in[6];
    const float* b1     = (const float*)d_in[7];
    const float* W2     = (const float*)d_in[8];
    const float* b2     = (const float*)d_in[9];
    const float* W3     = (const float*)d_in[10];
    float* out = (float*)d_out;
    const int N = in_sizes[0];           // 131072, divisible by PTSB=64
    const int blocks = N / PTSB;
    pinn_jet_wmma_kernel<<<blocks, WAVES * 32, 0, stream>>>(
        phi, theta, radius, hth, hph, brr, W1, b1, W2, b2, W3, out, N);
}